// TOMReorientation_3813930959208
// MI455X (gfx1250) — compile-verified
//
#include <hip/hip_runtime.h>

// Problem constants (fixed by the reference: flow is (2,3,96,96,96) f32)
#define DIM   96
#define D2    (DIM * DIM)          // 9216
#define D3    (DIM * DIM * DIM)    // 884736
#define BLK   (DIM * 2)            // 192 threads: (l, b) for one (i,j) row; 6 wave32s
#define MAX_ITERS 100
#define TOL   1e-5f

// Async-store builtin takes addrspace-qualified int* (probe-confirmed in round 1).
typedef __attribute__((address_space(1))) int i32_global;
typedef __attribute__((address_space(3))) int i32_lds;

#if defined(__gfx1250__) && __has_builtin(__builtin_amdgcn_global_store_async_from_lds_b32)
#define USE_ASYNC_STORE 1
#else
#define USE_ASYNC_STORE 0
#endif

// jnp.gradient semantics along one axis: central diff interior, one-sided edges.
__device__ __forceinline__ float grad_axis(const float* __restrict__ p,
                                           int idx, int coord, int stride) {
    if (coord == 0)       return p[idx + stride] - p[idx];
    if (coord == DIM - 1) return p[idx] - p[idx - stride];
    return 0.5f * (p[idx + stride] - p[idx - stride]);
}

__global__ __launch_bounds__(BLK)
void polar3x3_kernel(const float* __restrict__ flow, float* __restrict__ out) {
    __shared__ __align__(16) float sm[BLK * 9];  // 6912 B output staging

    const int tid = threadIdx.x;
    const int l   = tid >> 1;            // voxel x (fastest), per lane
    const int b   = tid & 1;             // batch
    const int j   = blockIdx.x;          // uniform -> scalar edge branches
    const int i   = blockIdx.y;          // uniform -> scalar edge branches
    const int idx = i * D2 + j * DIM + l;

    // ---- Build J + I : x[k*3+c] = d flow[b,c]/d axis_k  (k: 0=i,1=j,2=l) ----
    float x[9];
#pragma unroll
    for (int c = 0; c < 3; ++c) {
        const float* p = flow + (size_t)(b * 3 + c) * D3;
        __builtin_prefetch(p + idx, 0, 3);       // global_prefetch_b8
        x[0 * 3 + c] = grad_axis(p, idx, i, D2);
        x[1 * 3 + c] = grad_axis(p, idx, j, DIM);
        x[2 * 3 + c] = grad_axis(p, idx, l, 1);
    }
    x[0] += 1.0f; x[4] += 1.0f; x[8] += 1.0f;

    // ---- Newton polar iteration X <- 0.5*(X + X^{-T}), per-lane early exit ----
    // Matches reference masking: body runs, THEN mask = |tn - on| > TOL.
    float tn = 1.0f;
#pragma unroll 1
    for (int it = 0; it < MAX_ITERS; ++it) {
        const float on = tn;
        tn = sqrtf(x[0]*x[0] + x[1]*x[1] + x[2]*x[2] +
                   x[3]*x[3] + x[4]*x[4] + x[5]*x[5] +
                   x[6]*x[6] + x[7]*x[7] + x[8]*x[8]);

        // Cofactor matrix: inv(X)^T = cof(X) / det  (no transpose needed)
        const float c0 = x[4]*x[8] - x[5]*x[7];
        const float c1 = x[5]*x[6] - x[3]*x[8];
        const float c2 = x[3]*x[7] - x[4]*x[6];
        const float c3 = x[2]*x[7] - x[1]*x[8];
        const float c4 = x[0]*x[8] - x[2]*x[6];
        const float c5 = x[1]*x[6] - x[0]*x[7];
        const float c6 = x[1]*x[5] - x[2]*x[4];
        const float c7 = x[2]*x[3] - x[0]*x[5];
        const float c8 = x[0]*x[4] - x[1]*x[3];
        const float det = x[0]*c0 + x[1]*c1 + x[2]*c2;
        const float r   = 1.0f / det;

        x[0] = 0.5f * (x[0] + c0 * r);
        x[1] = 0.5f * (x[1] + c1 * r);
        x[2] = 0.5f * (x[2] + c2 * r);
        x[3] = 0.5f * (x[3] + c3 * r);
        x[4] = 0.5f * (x[4] + c4 * r);
        x[5] = 0.5f * (x[5] + c5 * r);
        x[6] = 0.5f * (x[6] + c6 * r);
        x[7] = 0.5f * (x[7] + c7 * r);
        x[8] = 0.5f * (x[8] + c8 * r);

        if (fabsf(tn - on) <= TOL) break;
    }

    // ---- Stage to LDS in output order (stride 9, gcd(9,32)=1 -> conflict-free) ----
    // Output element for this thread: out[m*9 + q], m = (i*96 + j)*192 + tid.
#pragma unroll
    for (int q = 0; q < 9; ++q) sm[tid * 9 + q] = x[q];
    __syncthreads();

    // ---- Coalesced writeback: 1728 contiguous floats per block ----
    float* gbase = out + (size_t)(i * DIM + j) * (BLK * 9);
#pragma unroll
    for (int k = 0; k < 9; ++k) {
        const int t = tid + k * BLK;
#if USE_ASYNC_STORE
        // GLOBAL_STORE_ASYNC_FROM_LDS_B32: LDS -> global via async DMA path,
        // tracked by ASYNCcnt, no data VGPRs consumed.
        __builtin_amdgcn_global_store_async_from_lds_b32(
            (i32_global*)(gbase + t), (i32_lds*)&sm[t], 0, 0);
#else
        gbase[t] = sm[t];
#endif
    }

#if USE_ASYNC_STORE
#if __has_builtin(__builtin_amdgcn_s_wait_asynccnt)
    __builtin_amdgcn_s_wait_asynccnt(0);
#else
    asm volatile("s_wait_asynccnt 0" ::: "memory");
#endif
#endif
}

extern "C" void kernel_launch(void* const* d_in, const int* in_sizes, int n_in,
                              void* d_out, int out_size, void* d_ws, size_t ws_size,
                              hipStream_t stream) {
    (void)n_in; (void)d_ws; (void)ws_size; (void)out_size; (void)in_sizes;
    const float* flow = (const float*)d_in[0];
    float* out = (float*)d_out;

    dim3 grid(DIM, DIM);   // (j, i): 9216 blocks x 192 threads = 1,769,472 matrices
    polar3x3_kernel<<<grid, BLK, 0, stream>>>(flow, out);
}